// MLLoss_16913581212137
// MI455X (gfx1250) — compile-verified
//
#include <hip/hip_runtime.h>

// ---------------------------------------------------------------------------
// Memory-bound hinge-loss reduction for MI455X (gfx1250, wave32).
//   traffic: 256MiB dist (f32 [B,16]) + 2*16MiB int32 labels -> ~13us @ 23.3TB/s
//   pass 1: grid-stride streaming (NT b128 loads), 4x unrolled w/ independent
//           accumulators, fully branchless per-row math, wave reduction via
//           V_WMMA_F32_16X16X4_F32 (B = ones), block partial to d_ws.
//   pass 2: single block reduces partials deterministically, writes mean.
// ---------------------------------------------------------------------------

typedef float vf4 __attribute__((ext_vector_type(4)));
typedef float v2f __attribute__((ext_vector_type(2)));
typedef float v8f __attribute__((ext_vector_type(8)));

__device__ __forceinline__ float hinge(float x) {
    return fmaxf(1.0f - x, 0.0f);
}

// Branchless per-row loss (doctor in {0,1,2}, real in {0,1}).
__device__ __forceinline__ float row_loss(vf4 q0, vf4 q1, vf4 q2, vf4 q3,
                                          int dlab, int rlab) {
    float h0 = hinge(q0.x);
    float h1 = hinge(q0.y);
    // balanced-tree sum of all 16 hinges (short VALU dependency chain)
    float t0 = (h0 + h1)            + (hinge(q0.z) + hinge(q0.w));
    float t1 = (hinge(q1.x) + hinge(q1.y)) + (hinge(q1.z) + hinge(q1.w));
    float t2 = (hinge(q2.x) + hinge(q2.y)) + (hinge(q2.z) + hinge(q2.w));
    float t3 = (hinge(q3.x) + hinge(q3.y)) + (hinge(q3.z) + hinge(q3.w));
    float hs = (t0 + t1) + (t2 + t3);

    float loss_c = q0.x + (hs - h0);                 // doctor == control
    float loss_p = q0.y + (hs - h1);                 // doctor == parkinson
    float loss_u = (rlab == 0) ? h1 : h0;            // doctor == unknown
    float loss   = (dlab == 1) ? loss_p : loss_u;    // v_cndmask chain,
    loss         = (dlab == 0) ? loss_c : loss;      // no branches
    return loss;
}

__global__ __launch_bounds__(256) void mlloss_partial_kernel(
    const vf4* __restrict__ dist,      // [B, 4] float4 = [B,16] f32
    const int* __restrict__ doctor,    // [B]
    const int* __restrict__ realv,     // [B]
    float* __restrict__ partials,      // [gridDim.x]
    int nrows)
{
    const int gtid   = blockIdx.x * blockDim.x + threadIdx.x;
    const int stride = gridDim.x * blockDim.x;

    float acc0 = 0.0f, acc1 = 0.0f, acc2 = 0.0f, acc3 = 0.0f;

    // main loop: 4 rows per iteration, all 24 loads issued before any wait
    int r = gtid;
    for (; r + 3 * stride < nrows; r += 4 * stride) {
        const int ra = r, rb = r + stride, rc = r + 2 * stride, rd = r + 3 * stride;
        const vf4* pa = dist + (size_t)ra * 4;
        const vf4* pb = dist + (size_t)rb * 4;
        const vf4* pc = dist + (size_t)rc * 4;
        const vf4* pd = dist + (size_t)rd * 4;

        vf4 a0 = __builtin_nontemporal_load(pa + 0);
        vf4 a1 = __builtin_nontemporal_load(pa + 1);
        vf4 a2 = __builtin_nontemporal_load(pa + 2);
        vf4 a3 = __builtin_nontemporal_load(pa + 3);
        vf4 b0 = __builtin_nontemporal_load(pb + 0);
        vf4 b1 = __builtin_nontemporal_load(pb + 1);
        vf4 b2 = __builtin_nontemporal_load(pb + 2);
        vf4 b3 = __builtin_nontemporal_load(pb + 3);
        vf4 c0 = __builtin_nontemporal_load(pc + 0);
        vf4 c1 = __builtin_nontemporal_load(pc + 1);
        vf4 c2 = __builtin_nontemporal_load(pc + 2);
        vf4 c3 = __builtin_nontemporal_load(pc + 3);
        vf4 d0 = __builtin_nontemporal_load(pd + 0);
        vf4 d1 = __builtin_nontemporal_load(pd + 1);
        vf4 d2 = __builtin_nontemporal_load(pd + 2);
        vf4 d3 = __builtin_nontemporal_load(pd + 3);

        int da = __builtin_nontemporal_load(doctor + ra);
        int db = __builtin_nontemporal_load(doctor + rb);
        int dc = __builtin_nontemporal_load(doctor + rc);
        int dd = __builtin_nontemporal_load(doctor + rd);
        int ea = __builtin_nontemporal_load(realv + ra);
        int eb = __builtin_nontemporal_load(realv + rb);
        int ec = __builtin_nontemporal_load(realv + rc);
        int ed = __builtin_nontemporal_load(realv + rd);

        acc0 += row_loss(a0, a1, a2, a3, da, ea);
        acc1 += row_loss(b0, b1, b2, b3, db, eb);
        acc2 += row_loss(c0, c1, c2, c3, dc, ec);
        acc3 += row_loss(d0, d1, d2, d3, dd, ed);
    }
    // remainder
    for (; r < nrows; r += stride) {
        const vf4* p = dist + (size_t)r * 4;
        vf4 q0 = __builtin_nontemporal_load(p + 0);
        vf4 q1 = __builtin_nontemporal_load(p + 1);
        vf4 q2 = __builtin_nontemporal_load(p + 2);
        vf4 q3 = __builtin_nontemporal_load(p + 3);
        int dl = __builtin_nontemporal_load(doctor + r);
        int rl = __builtin_nontemporal_load(realv + r);
        acc0 += row_loss(q0, q1, q2, q3, dl, rl);
    }

    float acc = (acc0 + acc1) + (acc2 + acc3);

    // ---- wave32 reduction with V_WMMA_F32_16X16X4_F32 ----
    // A: VGPR0 = per-lane partial, VGPR1 = 0
    //    => A[m][0] = p[m] (lanes 0-15), A[m][2] = p[m+16] (lanes 16-31),
    //       A[m][1] = A[m][3] = 0.
    // B: all ones  => D[m][n] = p[m] + p[m+16] for every n.
    // D VGPR j: lanes 0-15 = p[j]+p[j+16], lanes 16-31 = p[j+8]+p[j+24].
    v2f av; av[0] = acc;  av[1] = 0.0f;
    v2f bv; bv[0] = 1.0f; bv[1] = 1.0f;
    v8f cv = {};
    v8f dv = __builtin_amdgcn_wmma_f32_16x16x4_f32(
        /*neg_a=*/false, av, /*neg_b=*/false, bv,
        /*c_mod=*/(short)0, cv, /*reuse_a=*/false, /*reuse_b=*/false);

    // lane-local sum of 8 D VGPRs: lanes 0-15 hold sum(p[0..7]+p[16..23]),
    // lanes 16-31 hold sum(p[8..15]+p[24..31]); xor-16 completes the wave sum.
    float s = ((dv[0] + dv[1]) + (dv[2] + dv[3]))
            + ((dv[4] + dv[5]) + (dv[6] + dv[7]));
    s += __shfl_xor(s, 16, 32);

    __shared__ float wsum[8];
    const int lane = threadIdx.x & 31;
    const int wave = threadIdx.x >> 5;
    if (lane == 0) wsum[wave] = s;
    __syncthreads();
    if (threadIdx.x == 0) {
        float t = 0.0f;
        #pragma unroll
        for (int w = 0; w < 8; ++w) t += wsum[w];
        partials[blockIdx.x] = t;
    }
}

__global__ __launch_bounds__(256) void mlloss_final_kernel(
    const float* __restrict__ partials, int n,
    float* __restrict__ out, float inv_b)
{
    __shared__ float lds[256];
    float acc = 0.0f;
    for (int i = threadIdx.x; i < n; i += 256) acc += partials[i];
    lds[threadIdx.x] = acc;
    __syncthreads();
    #pragma unroll
    for (int s = 128; s > 0; s >>= 1) {
        if ((int)threadIdx.x < s) lds[threadIdx.x] += lds[threadIdx.x + s];
        __syncthreads();
    }
    if (threadIdx.x == 0) out[0] = lds[0] * inv_b;
}

extern "C" void kernel_launch(void* const* d_in, const int* in_sizes, int n_in,
                              void* d_out, int out_size, void* d_ws, size_t ws_size,
                              hipStream_t stream) {
    const vf4* dist   = (const vf4*)d_in[0];   // float32 [B,16]
    const int* doctor = (const int*)d_in[1];   // int     [B]
    const int* realv  = (const int*)d_in[2];   // int     [B]
    const int  nrows  = in_sizes[1];

    float* partials = (float*)d_ws;

    int grid = 4096;                            // 1M threads, 4 rows/thread at B=4M
    int max_grid = (nrows + 255) / 256;
    if (grid > max_grid) grid = max_grid;

    mlloss_partial_kernel<<<grid, 256, 0, stream>>>(dist, doctor, realv, partials, nrows);
    mlloss_final_kernel<<<1, 256, 0, stream>>>(partials, grid, (float*)d_out,
                                               1.0f / (float)nrows);
}